// AttentionEBM_38903813767558
// MI455X (gfx1250) — compile-verified
//
#include <hip/hip_runtime.h>
#include <cmath>

// ---------------------------------------------------------------------------
// AttentionEBM forward for MI455X (gfx1250), fp32 WMMA (V_WMMA_F32_16X16X4_F32)
// B=32, N=4096 (64x64 grid), OUT_DIM=512, K=512 gathered keys, H=64
// ---------------------------------------------------------------------------

typedef __attribute__((ext_vector_type(2))) float v2f;
typedef __attribute__((ext_vector_type(8))) float v8f;

__device__ __forceinline__ v8f wmma4(v2f a, v2f b, v8f c) {
  // D(16x16,f32) = A(16x4,f32) * B(4x16,f32) + C
  return __builtin_amdgcn_wmma_f32_16x16x4_f32(
      /*neg_a=*/false, a, /*neg_b=*/false, b,
      /*c_mod=*/(short)0, c, /*reuse_a=*/false, /*reuse_b=*/false);
}

__device__ __forceinline__ v8f vzero8() {
  v8f z = {0.f, 0.f, 0.f, 0.f, 0.f, 0.f, 0.f, 0.f};
  return z;
}

__device__ __forceinline__ float swishf(float v) {
  // v * sigmoid(v), rcp via v_rcp_f32 (1 ulp) instead of IEEE divide
  return v * __builtin_amdgcn_rcpf(1.0f + __expf(-v));
}

// acc[4] (16x64 C tile) += A(16xK, LDS row-major, stride lda) * W(Kx64 row-major)
__device__ __forceinline__ void gemm16(const float* As, int lda, int K,
                                       const float* W, v8f* acc) {
  const int lane = threadIdx.x & 31;
  const int m  = lane & 15;
  const int hi = (lane >> 4) & 1;
  for (int kb = 0; kb < K; kb += 4) {
    v2f a;
    a.x = As[m * lda + kb + 2 * hi];
    a.y = As[m * lda + kb + 2 * hi + 1];
#pragma unroll
    for (int nt = 0; nt < 4; ++nt) {
      const int n = m + nt * 16;
      v2f bv;
      bv.x = W[(kb + 2 * hi) * 64 + n];
      bv.y = W[(kb + 2 * hi + 1) * 64 + n];
      acc[nt] = wmma4(a, bv, acc[nt]);
    }
  }
}

// write C tile (bias + optional swish) to LDS row-major with given stride
__device__ __forceinline__ void epi_lds(float* S, int stride, v8f* acc,
                                        const float* bias, bool dosw) {
  const int lane = threadIdx.x & 31;
  const int m  = lane & 15;
  const int hi = (lane >> 4) & 1;
#pragma unroll
  for (int nt = 0; nt < 4; ++nt) {
    const int col = m + nt * 16;
    const float bv = bias ? bias[col] : 0.0f;
#pragma unroll
    for (int r = 0; r < 8; ++r) {
      float v = acc[nt][r] + bv;
      if (dosw) v = swishf(v);
      S[(r + 8 * hi) * stride + col] = v;
    }
  }
}

// write C tile (+bias) to global, row stride 64
__device__ __forceinline__ void epi_global(float* G, v8f* acc, const float* bias) {
  const int lane = threadIdx.x & 31;
  const int m  = lane & 15;
  const int hi = (lane >> 4) & 1;
#pragma unroll
  for (int nt = 0; nt < 4; ++nt) {
    const int col = m + nt * 16;
    const float bv = bias[col];
#pragma unroll
    for (int r = 0; r < 8; ++r) {
      G[(size_t)(r + 8 * hi) * 64 + col] = acc[nt][r] + bv;
    }
  }
}

// posenc(grid)[n][c2]  (c2 in [0,42))  -- used only by the one-shot init kernel
__device__ __forceinline__ float pos_pe_val(int n, int c2) {
  const float fi = (float)(n >> 6) * 0.015625f;
  const float fj = (float)(n & 63) * 0.015625f;
  if (c2 == 0) return fi;
  if (c2 == 1) return fj;
  const int cc = c2 - 2;
  const int f = cc >> 2;
  const int rem = cc & 3;
  const float base = (rem & 1) ? fj : fi;
  const float ang = base * (float)(1 << f);
  return (rem < 2) ? sinf(ang) : cosf(ang);
}

// posenc(out)[o][c2]   (c2 in [0,21))  -- used only by the one-shot init kernel
__device__ __forceinline__ float out_pe_val(int o, int c2) {
  const float v = (float)o * 0.015625f;
  if (c2 == 0) return v;
  const int cc = c2 - 1;
  const int f = cc >> 1;
  const float ang = v * (float)(1 << f);
  return ((cc & 1) == 0) ? sinf(ang) : cosf(ang);
}

// ---------------------------------------------------------------------------
// Kernel 0: batch-invariant posenc tables (accurate trig, one shot)
// pos_pe_t: (4096, 42), out_pe_t: (512, 21). grid = 714 * 256 = exact cover.
// ---------------------------------------------------------------------------
__global__ __launch_bounds__(256) void k_init(float* __restrict__ pos_pe_t,
                                              float* __restrict__ out_pe_t) {
  const int t = blockIdx.x * 256 + threadIdx.x;
  if (t < 4096 * 42) {
    pos_pe_t[t] = pos_pe_val(t / 42, t % 42);
  } else {
    const int u = t - 4096 * 42;
    if (u < 512 * 21) out_pe_t[u] = out_pe_val(u / 21, u % 21);
  }
}

// ---------------------------------------------------------------------------
// Kernel 1: pos branch -> pe (B,4096,64), pos_val (B,4096,64)
// grid = 32 * 64 blocks, 128 threads (4 waves x 16 rows)
// ---------------------------------------------------------------------------
__global__ __launch_bounds__(128) void k_pos(
    const float* __restrict__ x, const float* __restrict__ ilw,
    const float* __restrict__ ilb, const float* __restrict__ pos_pe_t,
    const float* __restrict__ w1, const float* __restrict__ b1,
    const float* __restrict__ w2, const float* __restrict__ b2,
    const float* __restrict__ w3, const float* __restrict__ b3,
    float* __restrict__ pe, float* __restrict__ pos_val) {
  __shared__ float lds[4 * (16 * 84 + 16 * 64 + 16)];
  const int wave = threadIdx.x >> 5;
  const int lane = threadIdx.x & 31;
  float* A0 = lds + wave * (16 * 84 + 16 * 64 + 16);
  float* A1 = A0 + 16 * 84;
  float* Xr = A1 + 16 * 64;
  const int b  = blockIdx.x >> 6;
  const int nb = blockIdx.x & 63;
  const int n0 = nb * 64 + wave * 16;

  if (lane < 16) Xr[lane] = x[(size_t)b * 4608 + 512 + n0 + lane];

  // build pos_embed tile (16 x 84) into LDS
  for (int t = lane; t < 16 * 84; t += 32) {
    const int row = t / 84;
    const int c = t % 84;
    float v;
    if (c < 42) {
      v = Xr[row] * ilw[c] + ilb[c];
    } else {
      v = pos_pe_t[(n0 + row) * 42 + (c - 42)];
    }
    A0[row * 84 + c] = v;
  }

  v8f acc[4];
#pragma unroll
  for (int i = 0; i < 4; ++i) acc[i] = vzero8();
  gemm16(A0, 84, 84, w1, acc);          // pos_embed @ inp_fc1
  epi_lds(A1, 64, acc, b1, true);       // swish

#pragma unroll
  for (int i = 0; i < 4; ++i) acc[i] = vzero8();
  gemm16(A1, 64, 64, w2, acc);          // @ inp_fc2  -> pe (raw)
  {
    const int m = lane & 15;
    const int hi = (lane >> 4) & 1;
#pragma unroll
    for (int nt = 0; nt < 4; ++nt) {
      const int col = m + nt * 16;
      const float bv = b2[col];
#pragma unroll
      for (int r = 0; r < 8; ++r) {
        const int row = r + 8 * hi;
        const float v = acc[nt][r] + bv;
        pe[((size_t)b * 4096 + n0 + row) * 64 + col] = v;
        A1[row * 64 + col] = swishf(v);
      }
    }
  }

#pragma unroll
  for (int i = 0; i < 4; ++i) acc[i] = vzero8();
  gemm16(A1, 64, 64, w3, acc);          // swish(pe) @ inp_fc3
  epi_global(pos_val + ((size_t)b * 4096 + n0) * 64, acc, b3);
}

// ---------------------------------------------------------------------------
// Kernel 2: out branch -> oe (B,512,64), out_val (B,512,64)
// grid = 32 * 8 blocks, 128 threads
// ---------------------------------------------------------------------------
__global__ __launch_bounds__(128) void k_out(
    const float* __restrict__ x, const float* __restrict__ olw,
    const float* __restrict__ olb, const float* __restrict__ out_pe_t,
    const float* __restrict__ w1, const float* __restrict__ b1,
    const float* __restrict__ w2, const float* __restrict__ b2,
    const float* __restrict__ w3, const float* __restrict__ b3,
    float* __restrict__ oe, float* __restrict__ out_val) {
  __shared__ float Wp[64 * 64];                      // out_fc1_w padded 63->64
  __shared__ float lds[4 * (16 * 64 + 16 * 64 + 16)];
  const int wave = threadIdx.x >> 5;
  const int lane = threadIdx.x & 31;
  float* A0 = lds + wave * (16 * 64 + 16 * 64 + 16);
  float* A1 = A0 + 16 * 64;
  float* Xr = A1 + 16 * 64;
  const int b  = blockIdx.x >> 3;
  const int ob = blockIdx.x & 7;
  const int o0 = ob * 64 + wave * 16;

  for (int t = threadIdx.x; t < 64 * 64; t += 128) {
    const int row = t >> 6;
    Wp[t] = (row < 63) ? w1[t] : 0.0f;
  }

  if (lane < 16) Xr[lane] = x[(size_t)b * 4608 + o0 + lane];

  // out_embed tile (16 x 64, col 63 zero)
  for (int t = lane; t < 16 * 64; t += 32) {
    const int row = t >> 6;
    const int c = t & 63;
    float v;
    if (c < 42) {
      v = Xr[row] * olw[c] + olb[c];
    } else if (c < 63) {
      v = out_pe_t[(o0 + row) * 21 + (c - 42)];
    } else {
      v = 0.0f;
    }
    A0[t] = v;
  }
  __syncthreads();

  v8f acc[4];
#pragma unroll
  for (int i = 0; i < 4; ++i) acc[i] = vzero8();
  gemm16(A0, 64, 64, Wp, acc);          // out_embed @ out_fc1 (padded)
  epi_lds(A1, 64, acc, b1, true);

#pragma unroll
  for (int i = 0; i < 4; ++i) acc[i] = vzero8();
  gemm16(A1, 64, 64, w2, acc);          // -> oe (raw)
  {
    const int m = lane & 15;
    const int hi = (lane >> 4) & 1;
#pragma unroll
    for (int nt = 0; nt < 4; ++nt) {
      const int col = m + nt * 16;
      const float bv = b2[col];
#pragma unroll
      for (int r = 0; r < 8; ++r) {
        const int row = r + 8 * hi;
        const float v = acc[nt][r] + bv;
        oe[((size_t)b * 512 + o0 + row) * 64 + col] = v;
        A1[row * 64 + col] = swishf(v);
      }
    }
  }

#pragma unroll
  for (int i = 0; i < 4; ++i) acc[i] = vzero8();
  gemm16(A1, 64, 64, w3, acc);          // swish(oe) @ out_fc3
  epi_global(out_val + ((size_t)b * 512 + o0) * 64, acc, b3);
}

// ---------------------------------------------------------------------------
// Kernel 3: at branch (gather) -> at (B,512,64)
// grid = 32 * 8 blocks, 128 threads
// ---------------------------------------------------------------------------
__global__ __launch_bounds__(128) void k_at(
    const float* __restrict__ x, const int* __restrict__ idx,
    const float* __restrict__ ilw, const float* __restrict__ ilb,
    const float* __restrict__ pos_pe_t, const float* __restrict__ w1,
    const float* __restrict__ b1, const float* __restrict__ w2,
    const float* __restrict__ b2, float* __restrict__ at) {
  __shared__ float Wp[64 * 64];                      // at_fc1_w padded 63->64
  __shared__ float lds[4 * (16 * 64 + 16 * 64 + 16)];
  __shared__ int NrAll[4 * 16];
  const int wave = threadIdx.x >> 5;
  const int lane = threadIdx.x & 31;
  float* A0 = lds + wave * (16 * 64 + 16 * 64 + 16);
  float* A1 = A0 + 16 * 64;
  float* Xr = A1 + 16 * 64;
  int* Nr = NrAll + wave * 16;
  const int b  = blockIdx.x >> 3;
  const int kb = blockIdx.x & 7;
  const int k0 = kb * 64 + wave * 16;

  for (int t = threadIdx.x; t < 64 * 64; t += 128) {
    const int row = t >> 6;
    Wp[t] = (row < 63) ? w1[t] : 0.0f;
  }

  if (lane < 16) {
    const int n = idx[(size_t)b * 512 + k0 + lane];
    Nr[lane] = n;
    Xr[lane] = x[(size_t)b * 4608 + 512 + n];
  }

  // gathered at_embed tile (16 x 64, col 63 zero)
  for (int t = lane; t < 16 * 64; t += 32) {
    const int row = t >> 6;
    const int c = t & 63;
    float v;
    if (c < 42) {
      v = Xr[row] * ilw[c] + ilb[c];
    } else if (c < 63) {
      v = pos_pe_t[Nr[row] * 42 + (c - 42)];
    } else {
      v = 0.0f;
    }
    A0[t] = v;
  }
  __syncthreads();

  v8f acc[4];
#pragma unroll
  for (int i = 0; i < 4; ++i) acc[i] = vzero8();
  gemm16(A0, 64, 64, Wp, acc);          // at_embed @ at_fc1 (padded)
  epi_lds(A1, 64, acc, b1, true);

#pragma unroll
  for (int i = 0; i < 4; ++i) acc[i] = vzero8();
  gemm16(A1, 64, 64, w2, acc);          // @ at_fc2
  epi_global(at + ((size_t)b * 512 + k0) * 64, acc, b2);
}

// ---------------------------------------------------------------------------
// Flash-style online-softmax attention for one 16-query wave tile
// Km/Vm: (nk, 64) row-major. acc out = softmax(Aq @ Km^T) @ Vm (16 x 64)
// ---------------------------------------------------------------------------
__device__ void attend(const float* Aq, float* Ps, const float* __restrict__ Km,
                       const float* __restrict__ Vm, int nk, v8f* acc) {
  const int lane = threadIdx.x & 31;
  const int m  = lane & 15;
  const int hi = (lane >> 4) & 1;
  float mrow[8], lrow[8];
#pragma unroll
  for (int r = 0; r < 8; ++r) { mrow[r] = -1e30f; lrow[r] = 0.0f; }
#pragma unroll
  for (int i = 0; i < 4; ++i) acc[i] = vzero8();

  for (int kc = 0; kc < nk; kc += 64) {
    if (kc + 64 < nk) {  // prefetch next key/value chunk into caches
      __builtin_prefetch(Km + (size_t)(kc + 64 + m) * 64, 0, 3);
      __builtin_prefetch(Vm + (size_t)(kc + 64 + m) * 64, 0, 3);
    }
    // logits S = Aq(16x64) @ Km[kc:kc+64]^T  (keys striped over lanes)
    v8f s[4];
#pragma unroll
    for (int i = 0; i < 4; ++i) s[i] = vzero8();
    for (int kk = 0; kk < 64; kk += 4) {
      v2f a;
      a.x = Aq[m * 64 + kk + 2 * hi];
      a.y = Aq[m * 64 + kk + 2 * hi + 1];
#pragma unroll
      for (int nt = 0; nt < 4; ++nt) {
        const int key = kc + m + nt * 16;
        v2f bv;
        bv.x = Km[(size_t)key * 64 + kk + 2 * hi];
        bv.y = Km[(size_t)key * 64 + kk + 2 * hi + 1];
        s[nt] = wmma4(a, bv, s[nt]);
      }
    }
    // online softmax per C-row (16 lanes of each half hold one row)
#pragma unroll
    for (int r = 0; r < 8; ++r) {
      float cm = fmaxf(fmaxf(s[0][r], s[1][r]), fmaxf(s[2][r], s[3][r]));
      cm = fmaxf(cm, __shfl_xor(cm, 1, 32));
      cm = fmaxf(cm, __shfl_xor(cm, 2, 32));
      cm = fmaxf(cm, __shfl_xor(cm, 4, 32));
      cm = fmaxf(cm, __shfl_xor(cm, 8, 32));
      const float nm = fmaxf(mrow[r], cm);
      const float sc = __expf(mrow[r] - nm);
      mrow[r] = nm;
      float ps = 0.0f;
#pragma unroll
      for (int nt = 0; nt < 4; ++nt) {
        const float p = __expf(s[nt][r] - nm);
        s[nt][r] = p;
        ps += p;
      }
      ps += __shfl_xor(ps, 1, 32);
      ps += __shfl_xor(ps, 2, 32);
      ps += __shfl_xor(ps, 4, 32);
      ps += __shfl_xor(ps, 8, 32);
      lrow[r] = lrow[r] * sc + ps;
#pragma unroll
      for (int nt = 0; nt < 4; ++nt) acc[nt][r] *= sc;
    }
    // P -> LDS (A layout round-trip), then acc += P @ V
#pragma unroll
    for (int nt = 0; nt < 4; ++nt)
#pragma unroll
      for (int r = 0; r < 8; ++r)
        Ps[(r + 8 * hi) * 64 + m + nt * 16] = s[nt][r];
    for (int kk = 0; kk < 64; kk += 4) {
      v2f a;
      a.x = Ps[m * 64 + kk + 2 * hi];
      a.y = Ps[m * 64 + kk + 2 * hi + 1];
#pragma unroll
      for (int nt = 0; nt < 4; ++nt) {
        const int n = m + nt * 16;
        v2f bv;
        bv.x = Vm[(size_t)(kc + kk + 2 * hi) * 64 + n];
        bv.y = Vm[(size_t)(kc + kk + 2 * hi + 1) * 64 + n];
        acc[nt] = wmma4(a, bv, acc[nt]);
      }
    }
  }
  // normalize (rcp instead of divide)
#pragma unroll
  for (int r = 0; r < 8; ++r) {
    const float rl = __builtin_amdgcn_rcpf(lrow[r]);
#pragma unroll
    for (int nt = 0; nt < 4; ++nt) acc[nt][r] *= rl;
  }
}

// ---------------------------------------------------------------------------
// Kernel 4: dual attention + final MLP -> out (B,512)
// grid = 32 * 16 blocks, 64 threads (2 waves x 16 queries)
// ---------------------------------------------------------------------------
__global__ __launch_bounds__(64) void k_attn(
    const float* __restrict__ at, const float* __restrict__ oe,
    const float* __restrict__ pe, const float* __restrict__ out_val,
    const float* __restrict__ pos_val, const float* __restrict__ fc1w,
    const float* __restrict__ fc1b, const float* __restrict__ fc2w,
    const float* __restrict__ fc2b, const float* __restrict__ fc3w,
    const float* __restrict__ fc3b, float* __restrict__ out) {
  __shared__ float lds[2 * (16 * 64 + 16 * 64 + 16 * 192)];
  const int wave = threadIdx.x >> 5;
  const int lane = threadIdx.x & 31;
  float* Aq = lds + wave * (16 * 64 + 16 * 64 + 16 * 192);
  float* Ps = Aq + 16 * 64;
  float* Es = Ps + 16 * 64;
  const int b  = blockIdx.x >> 4;
  const int qt = ((blockIdx.x & 15) << 1) + wave;
  const int q0 = qt * 16;

  // query tile (at) -> LDS
  for (int t = lane; t < 16 * 64; t += 32)
    Aq[t] = at[((size_t)b * 512 + q0 + (t >> 6)) * 64 + (t & 63)];

  v8f accO[4], accP[4];
  attend(Aq, Ps, oe + (size_t)b * 512 * 64, out_val + (size_t)b * 512 * 64, 512, accO);
  attend(Aq, Ps, pe + (size_t)b * 4096 * 64, pos_val + (size_t)b * 4096 * 64, 4096, accP);

  // emb = [at | pos_agg | out_agg] (16 x 192) in LDS
  for (int t = lane; t < 16 * 64; t += 32)
    Es[(t >> 6) * 192 + (t & 63)] = Aq[t];
  epi_lds(Es + 64, 192, accP, nullptr, false);
  epi_lds(Es + 128, 192, accO, nullptr, false);

  v8f acc[4];
#pragma unroll
  for (int i = 0; i < 4; ++i) acc[i] = vzero8();
  gemm16(Es, 192, 192, fc1w, acc);      // emb @ fc1
  epi_lds(Ps, 64, acc, fc1b, true);

#pragma unroll
  for (int i = 0; i < 4; ++i) acc[i] = vzero8();
  gemm16(Ps, 64, 64, fc2w, acc);        // @ fc2

  // h2 = swish(acc + fc2b); out = h2 @ fc3 + b  (row dot via shuffles)
  const int m  = lane & 15;
  const int hi = (lane >> 4) & 1;
#pragma unroll
  for (int r = 0; r < 8; ++r) {
    float dot = 0.0f;
#pragma unroll
    for (int nt = 0; nt < 4; ++nt) {
      const int col = m + nt * 16;
      const float v = swishf(acc[nt][r] + fc2b[col]);
      dot += v * fc3w[col];
    }
    dot += __shfl_xor(dot, 1, 32);
    dot += __shfl_xor(dot, 2, 32);
    dot += __shfl_xor(dot, 4, 32);
    dot += __shfl_xor(dot, 8, 32);
    if (m == 0)
      out[(size_t)b * 512 + q0 + r + 8 * hi] = dot + fc3b[0];
  }
}

// ---------------------------------------------------------------------------
extern "C" void kernel_launch(void* const* d_in, const int* in_sizes, int n_in,
                              void* d_out, int out_size, void* d_ws,
                              size_t ws_size, hipStream_t stream) {
  (void)in_sizes; (void)n_in; (void)out_size; (void)ws_size;
  const float* x   = (const float*)d_in[0];
  const int*   idx = (const int*)d_in[1];
  const float* inp_linear_w = (const float*)d_in[2];
  const float* inp_linear_b = (const float*)d_in[3];
  const float* out_linear_w = (const float*)d_in[4];
  const float* out_linear_b = (const float*)d_in[5];
  const float* inp_fc1_w = (const float*)d_in[6];
  const float* inp_fc1_b = (const float*)d_in[7];
  const float* inp_fc2_w = (const float*)d_in[8];
  const float* inp_fc2_b = (const float*)d_in[9];
  const float* inp_fc3_w = (const float*)d_in[10];
  const float* inp_fc3_b = (const float*)d_in[11];
  const float* out_fc1_w = (const float*)d_in[12];
  const float* out_fc1_b = (const float*)d_in[13];
  const float* out_fc2_w = (const float*)d_in[14];
  const float* out_fc2_b = (const float*)d_in[15];
  const float* out_fc3_w = (const float*)d_in[16];
  const float* out_fc3_b = (const float*)d_in[17];
  const float* at_fc1_w = (const float*)d_in[18];
  const float* at_fc1_b = (const float*)d_in[19];
  const float* at_fc2_w = (const float*)d_in[20];
  const float* at_fc2_b = (const float*)d_in[21];
  const float* fc1_w = (const float*)d_in[22];
  const float* fc1_b = (const float*)d_in[23];
  const float* fc2_w = (const float*)d_in[24];
  const float* fc2_b = (const float*)d_in[25];
  const float* fc3_w = (const float*)d_in[26];
  const float* fc3_b = (const float*)d_in[27];

  float* ws = (float*)d_ws;
  float* pe       = ws;                                   // 32*4096*64
  float* pos_val  = pe + (size_t)32 * 4096 * 64;          // 32*4096*64
  float* oe       = pos_val + (size_t)32 * 4096 * 64;     // 32*512*64
  float* out_val  = oe + (size_t)32 * 512 * 64;           // 32*512*64
  float* at       = out_val + (size_t)32 * 512 * 64;      // 32*512*64
  float* pos_pe_t = at + (size_t)32 * 512 * 64;           // 4096*42
  float* out_pe_t = pos_pe_t + (size_t)4096 * 42;         // 512*21
  float* out = (float*)d_out;

  k_init<<<714, 256, 0, stream>>>(pos_pe_t, out_pe_t);
  k_pos<<<32 * 64, 128, 0, stream>>>(x, inp_linear_w, inp_linear_b, pos_pe_t,
                                     inp_fc1_w, inp_fc1_b, inp_fc2_w, inp_fc2_b,
                                     inp_fc3_w, inp_fc3_b, pe, pos_val);
  k_out<<<32 * 8, 128, 0, stream>>>(x, out_linear_w, out_linear_b, out_pe_t,
                                    out_fc1_w, out_fc1_b, out_fc2_w, out_fc2_b,
                                    out_fc3_w, out_fc3_b, oe, out_val);
  k_at<<<32 * 8, 128, 0, stream>>>(x, idx, inp_linear_w, inp_linear_b, pos_pe_t,
                                   at_fc1_w, at_fc1_b, at_fc2_w, at_fc2_b, at);
  k_attn<<<32 * 16, 64, 0, stream>>>(at, oe, pe, out_val, pos_val,
                                     fc1_w, fc1_b, fc2_w, fc2_b, fc3_w, fc3_b,
                                     out);
}